// GrangerGPVAR_75874892251461
// MI455X (gfx1250) — compile-verified
//
#include <hip/hip_runtime.h>
#include <math.h>

// CDNA5 f32 WMMA fragment types: A/B = 2 VGPRs, C/D = 8 VGPRs (wave32)
typedef float v2f __attribute__((ext_vector_type(2)));
typedef float v8f __attribute__((ext_vector_type(8)));
typedef int gp_v4i __attribute__((vector_size(16)));  // b128 transfer unit

#define NB 64   // Cholesky / TRSM block size
#define LDP 68  // padded LDS row stride in dwords (conflict-free, 8B aligned)

#define GP_AS1 __attribute__((address_space(1)))
#define GP_AS3 __attribute__((address_space(3)))

#if __has_builtin(__builtin_amdgcn_global_load_async_to_lds_b128)
#define GP_ASYNC 1
#endif

__device__ __forceinline__ v8f wmma_f32_4(v2f a, v2f b, v8f c) {
  // D = A(16x4) * B(4x16) + C(16x16), all f32
  return __builtin_amdgcn_wmma_f32_16x16x4_f32(false, a, false, b, (short)0, c,
                                               false, false);
}

// Stage a 64x64 f32 tile (row-major, row stride gstride) into LDS with padded
// row stride LDP. 128 threads; 8 x b128 transfers per thread per panel.
__device__ __forceinline__ void gp_stage64(float* lds, const float* g,
                                           size_t gstride, int tid) {
#ifdef GP_ASYNC
  GP_AS3 float* lp = (GP_AS3 float*)lds;
  for (int c = tid; c < 64 * 16; c += 128) {
    int row = c >> 4, c4 = (c & 15) << 2;
    __builtin_amdgcn_global_load_async_to_lds_b128(
        (GP_AS1 gp_v4i*)(g + (size_t)row * gstride + c4),
        (GP_AS3 gp_v4i*)(lp + row * LDP + c4), 0, 0);
  }
#else
  for (int c = tid; c < 64 * 16; c += 128) {
    int row = c >> 4, c4 = (c & 15) << 2;
    float4 v = *(const float4*)(g + (size_t)row * gstride + c4);
    *(float4*)(lds + row * LDP + c4) = v;
  }
#endif
}

// Wait for this wave's async stages, then workgroup barrier.
__device__ __forceinline__ void gp_async_join() {
#ifdef GP_ASYNC
#if __has_builtin(__builtin_amdgcn_s_wait_asynccnt)
  __builtin_amdgcn_s_wait_asynccnt(0);
#else
  asm volatile("s_wait_asynccnt 0x0" ::: "memory");
#endif
#endif
  __syncthreads();
}

// ---------------------------------------------------------------------------
// Row squared-norms per source: out[s*count + i] = sum_k Z[s,i,k]^2
// ---------------------------------------------------------------------------
__global__ void gp_rownorm(const float* __restrict__ Z, float* __restrict__ out,
                           int total, int d) {
  int i = blockIdx.x * blockDim.x + threadIdx.x;
  if (i >= total) return;
  const float* p = Z + (size_t)i * d;
  float s = 0.f;
  for (int k = 0; k < d; ++k) s += p[k] * p[k];
  out[i] = s;
}

// ---------------------------------------------------------------------------
// Combined RBF kernel matrix build, one 16x16 tile per wave (32 threads).
// out[r, c] = sum_s W[s] * exp(-0.5 * (||zr||^2 + ||zc||^2 - 2 zr.zc) / ls2_s)
// ---------------------------------------------------------------------------
__global__ void gp_build_kernel_mat(
    const float* __restrict__ Zr, const float* __restrict__ Zc,
    const float* __restrict__ rnr, const float* __restrict__ rnc,
    const float* __restrict__ W, const float* __restrict__ log_ls,
    const float* __restrict__ log_noise, int add_diag, int S, int nr, int nc,
    int d, float* __restrict__ out) {
  const int lane = threadIdx.x & 31;
  const int half = lane >> 4;  // 0/1
  const int l16 = lane & 15;
  const int row0 = blockIdx.y * 16;
  const int col0 = blockIdx.x * 16;
  const int row = row0 + l16;  // A-fragment row held by this lane
  const int col = col0 + l16;  // B-fragment column held by this lane

  float accv[8] = {0.f, 0.f, 0.f, 0.f, 0.f, 0.f, 0.f, 0.f};

  for (int s = 0; s < S; ++s) {
    const float* za = Zr + ((size_t)s * nr + row) * d;
    const float* zb = Zc + ((size_t)s * nc + col) * d;
    v8f dot = {0.f, 0.f, 0.f, 0.f, 0.f, 0.f, 0.f, 0.f};
    if (d == 16) {
      // batch all 8 fragment loads, then 4 chained WMMAs
      v2f a[4], b[4];
#pragma unroll
      for (int ks = 0; ks < 4; ++ks) {
        a[ks] = *(const v2f*)(za + ks * 4 + 2 * half);
        b[ks] = *(const v2f*)(zb + ks * 4 + 2 * half);
      }
#pragma unroll
      for (int ks = 0; ks < 4; ++ks) dot = wmma_f32_4(a[ks], b[ks], dot);
    } else {
      for (int ks = 0; ks < (d >> 2); ++ks) {
        v2f a = *(const v2f*)(za + ks * 4 + 2 * half);
        v2f b = *(const v2f*)(zb + ks * 4 + 2 * half);
        dot = wmma_f32_4(a, b, dot);
      }
    }
    float ls = __expf(log_ls[s]);
    float ls2 = fmaxf(ls * ls, 1e-6f);
    float cfac = -0.5f / ls2;
    float w = W[s];
    float rb = rnc[(size_t)s * nc + col];
#pragma unroll
    for (int vr = 0; vr < 8; ++vr) {
      float ra = rnr[(size_t)s * nr + row0 + vr + 8 * half];
      float d2 = ra + rb - 2.0f * dot[vr];
      accv[vr] += w * __expf(cfac * d2);
    }
  }

  float dadd = 0.f;
  if (add_diag) dadd = __expf(log_noise[0]) + 1e-6f;  // noise + JITTER
#pragma unroll
  for (int vr = 0; vr < 8; ++vr) {
    int r = row0 + vr + 8 * half;
    float v = accv[vr];
    if (add_diag && r == col) v += dadd;
    out[(size_t)r * nc + col] = v;
  }
}

// ---------------------------------------------------------------------------
// Cholesky: unblocked factorization of the NB x NB diagonal block in LDS.
// ---------------------------------------------------------------------------
__global__ void gp_chol_diag(float* __restrict__ A, int n, int kb) {
  __shared__ float Ls[NB][NB + 1];
  int tid = threadIdx.x;
  for (int idx = tid; idx < NB * NB; idx += blockDim.x) {
    int i = idx / NB, j = idx % NB;
    Ls[i][j] = (j <= i) ? A[(size_t)(kb + i) * n + kb + j] : 0.f;
  }
  __syncthreads();
  for (int j = 0; j < NB; ++j) {
    if (tid == 0) Ls[j][j] = sqrtf(Ls[j][j]);
    __syncthreads();
    float inv = 1.0f / Ls[j][j];
    for (int i = j + 1 + tid; i < NB; i += blockDim.x) Ls[i][j] *= inv;
    __syncthreads();
    for (int idx = tid; idx < NB * NB; idx += blockDim.x) {
      int i = idx / NB, t = idx % NB;
      if (i > j && t > j && t <= i) Ls[i][t] -= Ls[i][j] * Ls[t][j];
    }
    __syncthreads();
  }
  for (int idx = tid; idx < NB * NB; idx += blockDim.x) {
    int i = idx / NB, j = idx % NB;
    if (j <= i) A[(size_t)(kb + i) * n + kb + j] = Ls[i][j];
  }
}

// ---------------------------------------------------------------------------
// Cholesky: panel TRSM. Each thread solves one row r against L11^-T.
// ---------------------------------------------------------------------------
__global__ void __launch_bounds__(128) gp_chol_trsm(float* __restrict__ A,
                                                    int n, int kb) {
  __shared__ float Ls[NB][NB + 1];
  __shared__ float Xs[NB][128];
  int tid = threadIdx.x;
  for (int idx = tid; idx < NB * NB; idx += blockDim.x) {
    int i = idx / NB, j = idx % NB;
    Ls[i][j] = (j <= i) ? A[(size_t)(kb + i) * n + kb + j] : 0.f;
  }
  __syncthreads();
  int r = kb + NB + blockIdx.x * blockDim.x + tid;
  if (r >= n) return;
  float* arow = A + (size_t)r * n + kb;
  for (int c = 0; c < NB; ++c) Xs[c][tid] = arow[c];
  for (int c = 0; c < NB; ++c) {
    float s = Xs[c][tid];
    for (int t = 0; t < c; ++t) s -= Ls[c][t] * Xs[t][tid];
    Xs[c][tid] = s / Ls[c][c];
  }
  for (int c = 0; c < NB; ++c) arow[c] = Xs[c][tid];
}

// ---------------------------------------------------------------------------
// Cholesky SYRK: A22 -= L21 * L21^T, 64x64 tile per 128-thread workgroup.
// Panels async-staged into LDS; each wave computes a 2x2 block of 16x16 WMMAs.
// ---------------------------------------------------------------------------
__global__ void __launch_bounds__(128) gp_chol_syrk(float* __restrict__ A,
                                                    int n, int kb) {
  if (blockIdx.x > blockIdx.y) return;  // uniform: keep lower 64x64 tiles
  __shared__ float Als[64][LDP];
  __shared__ float Bls[64][LDP];
  const int tid = threadIdx.x;
  const int base = kb + NB;
  const int rowblk = base + blockIdx.y * 64;
  const int colblk = base + blockIdx.x * 64;

  gp_stage64(&Als[0][0], A + (size_t)rowblk * n + kb, n, tid);
  gp_stage64(&Bls[0][0], A + (size_t)colblk * n + kb, n, tid);
  gp_async_join();

  const int wave = tid >> 5, lane = tid & 31;
  const int half = lane >> 4, l16 = lane & 15;
  const int sr = (wave >> 1) * 32, sc = (wave & 1) * 32;

  v8f acc00 = {0.f, 0.f, 0.f, 0.f, 0.f, 0.f, 0.f, 0.f};
  v8f acc01 = acc00, acc10 = acc00, acc11 = acc00;
#pragma unroll
  for (int ks = 0; ks < 64; ks += 4) {
    const int k = ks + 2 * half;
    v2f a0 = *(const v2f*)&Als[sr + l16][k];
    v2f a1 = *(const v2f*)&Als[sr + 16 + l16][k];
    v2f b0 = *(const v2f*)&Bls[sc + l16][k];
    v2f b1 = *(const v2f*)&Bls[sc + 16 + l16][k];
    acc00 = wmma_f32_4(a0, b0, acc00);
    acc01 = wmma_f32_4(a0, b1, acc01);
    acc10 = wmma_f32_4(a1, b0, acc10);
    acc11 = wmma_f32_4(a1, b1, acc11);
  }
#pragma unroll
  for (int vr = 0; vr < 8; ++vr) {
    int r0 = rowblk + sr + vr + 8 * half;
    int c0 = colblk + sc + l16;
    A[(size_t)r0 * n + c0] -= acc00[vr];
    A[(size_t)r0 * n + c0 + 16] -= acc01[vr];
    A[(size_t)(r0 + 16) * n + c0] -= acc10[vr];
    A[(size_t)(r0 + 16) * n + c0 + 16] -= acc11[vr];
  }
}

// ---------------------------------------------------------------------------
// Blocked forward substitution on V (n x m): GEMM update
// V[kb:kb+64, cb:cb+64] -= L[kb:kb+64, 0:kb] @ V[0:kb, cb:cb+64]
// K staged through LDS in 64-chunks; 2x2 WMMA register block per wave.
// ---------------------------------------------------------------------------
__global__ void __launch_bounds__(128) gp_panel_gemm(const float* __restrict__ L,
                                                     int n,
                                                     float* __restrict__ V,
                                                     int m, int kb) {
  __shared__ float Als[64][LDP];  // L rows (M x K chunk)
  __shared__ float Bls[64][LDP];  // V rows (K chunk x N)
  const int tid = threadIdx.x;
  const int colblk = blockIdx.x * 64;
  const int wave = tid >> 5, lane = tid & 31;
  const int half = lane >> 4, l16 = lane & 15;
  const int sr = (wave >> 1) * 32, sc = (wave & 1) * 32;

  v8f acc00 = {0.f, 0.f, 0.f, 0.f, 0.f, 0.f, 0.f, 0.f};
  v8f acc01 = acc00, acc10 = acc00, acc11 = acc00;

  for (int kc = 0; kc < kb; kc += 64) {
    gp_stage64(&Als[0][0], L + (size_t)kb * n + kc, n, tid);
    gp_stage64(&Bls[0][0], V + (size_t)kc * m + colblk, m, tid);
    gp_async_join();
#pragma unroll
    for (int ks = 0; ks < 64; ks += 4) {
      const int k = ks + 2 * half;
      v2f a0 = *(const v2f*)&Als[sr + l16][k];
      v2f a1 = *(const v2f*)&Als[sr + 16 + l16][k];
      v2f b0, b1;  // B[k][N]: column reads, consecutive lanes -> no conflicts
      b0.x = Bls[k][sc + l16];
      b0.y = Bls[k + 1][sc + l16];
      b1.x = Bls[k][sc + 16 + l16];
      b1.y = Bls[k + 1][sc + 16 + l16];
      acc00 = wmma_f32_4(a0, b0, acc00);
      acc01 = wmma_f32_4(a0, b1, acc01);
      acc10 = wmma_f32_4(a1, b0, acc10);
      acc11 = wmma_f32_4(a1, b1, acc11);
    }
    __syncthreads();  // all waves done reading before next stage overwrites
  }
#pragma unroll
  for (int vr = 0; vr < 8; ++vr) {
    int r0 = kb + sr + vr + 8 * half;
    int c0 = colblk + sc + l16;
    V[(size_t)r0 * m + c0] -= acc00[vr];
    V[(size_t)r0 * m + c0 + 16] -= acc01[vr];
    V[(size_t)(r0 + 16) * m + c0] -= acc10[vr];
    V[(size_t)(r0 + 16) * m + c0 + 16] -= acc11[vr];
  }
}

// Small triangular solve of the diagonal block against all m RHS columns.
__global__ void __launch_bounds__(128) gp_panel_trs(const float* __restrict__ L,
                                                    int n, float* __restrict__ V,
                                                    int m, int kb) {
  __shared__ float Ls[NB][NB + 1];
  __shared__ float Xs[NB][128];
  int tid = threadIdx.x;
  for (int idx = tid; idx < NB * NB; idx += blockDim.x) {
    int i = idx / NB, j = idx % NB;
    Ls[i][j] = (j <= i) ? L[(size_t)(kb + i) * n + kb + j] : 0.f;
  }
  __syncthreads();
  int c = blockIdx.x * blockDim.x + tid;
  if (c >= m) return;
  for (int r = 0; r < NB; ++r) Xs[r][tid] = V[(size_t)(kb + r) * m + c];
  for (int r = 0; r < NB; ++r) {
    float s = Xs[r][tid];
    for (int t = 0; t < r; ++t) s -= Ls[r][t] * Xs[t][tid];
    Xs[r][tid] = s / Ls[r][r];
  }
  for (int r = 0; r < NB; ++r) V[(size_t)(kb + r) * m + c] = Xs[r][tid];
}

// ---------------------------------------------------------------------------
// Single-RHS substitution (alpha solve), one workgroup, O(n^2) — negligible.
// ---------------------------------------------------------------------------
__global__ void gp_trsv_fwd(const float* __restrict__ L, int n,
                            const float* __restrict__ y, float* __restrict__ z) {
  __shared__ float red[256];
  int tid = threadIdx.x;
  for (int j = 0; j < n; ++j) {
    float s = 0.f;
    for (int t = tid; t < j; t += 256) s += L[(size_t)j * n + t] * z[t];
    red[tid] = s;
    __syncthreads();
    for (int off = 128; off > 0; off >>= 1) {
      if (tid < off) red[tid] += red[tid + off];
      __syncthreads();
    }
    if (tid == 0) z[j] = (y[j] - red[0]) / L[(size_t)j * n + j];
    __syncthreads();
  }
}

__global__ void gp_trsv_bwd(const float* __restrict__ L, int n,
                            const float* __restrict__ z,
                            float* __restrict__ alpha) {
  __shared__ float red[256];
  int tid = threadIdx.x;
  for (int jj = 0; jj < n; ++jj) {
    int j = n - 1 - jj;
    float s = 0.f;
    for (int t = j + 1 + tid; t < n; t += 256)
      s += L[(size_t)t * n + j] * alpha[t];
    red[tid] = s;
    __syncthreads();
    for (int off = 128; off > 0; off >>= 1) {
      if (tid < off) red[tid] += red[tid + off];
      __syncthreads();
    }
    if (tid == 0) alpha[j] = (z[j] - red[0]) / L[(size_t)j * n + j];
    __syncthreads();
  }
}

// ---------------------------------------------------------------------------
// mean[j] = sum_i KxT[i,j] * alpha[i]     (KxT = V before the solve)
// ---------------------------------------------------------------------------
__global__ void gp_mean(const float* __restrict__ KxT,
                        const float* __restrict__ alpha,
                        float* __restrict__ out, int n, int m) {
  int j = blockIdx.x * blockDim.x + threadIdx.x;
  if (j >= m) return;
  float s = 0.f;
  for (int i = 0; i < n; ++i) s += KxT[(size_t)i * m + j] * alpha[i];
  out[j] = s;
}

// var[j] = max(sum(W) - sum_i v[i,j]^2 + noise, 1e-9)
__global__ void gp_var(const float* __restrict__ V, const float* __restrict__ W,
                       const float* __restrict__ log_noise,
                       float* __restrict__ out, int n, int m, int S) {
  int j = blockIdx.x * blockDim.x + threadIdx.x;
  if (j >= m) return;
  float ss = 0.f;
  for (int i = 0; i < n; ++i) {
    float t = V[(size_t)i * m + j];
    ss += t * t;
  }
  float kss = 0.f;
  for (int s = 0; s < S; ++s) kss += W[s];
  float noise = __expf(log_noise[0]);
  out[j] = fmaxf(kss - ss + noise, 1e-9f);
}

// mll = -0.5*y.alpha - sum_i log(L[i,i]) - 0.5*n*log(2*pi)
__global__ void gp_mll(const float* __restrict__ L, int n,
                       const float* __restrict__ y,
                       const float* __restrict__ alpha,
                       float* __restrict__ out) {
  __shared__ float r1[256], r2[256];
  int tid = threadIdx.x;
  float s1 = 0.f, s2 = 0.f;
  for (int i = tid; i < n; i += 256) {
    s1 += y[i] * alpha[i];
    s2 += __logf(L[(size_t)i * n + i]);
  }
  r1[tid] = s1;
  r2[tid] = s2;
  __syncthreads();
  for (int off = 128; off > 0; off >>= 1) {
    if (tid < off) {
      r1[tid] += r1[tid + off];
      r2[tid] += r2[tid + off];
    }
    __syncthreads();
  }
  if (tid == 0)
    out[0] = -0.5f * r1[0] - r2[0] -
             0.5f * (float)n * logf(6.283185307179586f);
}

// ---------------------------------------------------------------------------
extern "C" void kernel_launch(void* const* d_in, const int* in_sizes, int n_in,
                              void* d_out, int out_size, void* d_ws,
                              size_t ws_size, hipStream_t stream) {
  (void)n_in;
  (void)out_size;
  (void)ws_size;
  const float* Zt = (const float*)d_in[0];   // (S, n, d)
  const float* Zs = (const float*)d_in[1];   // (S, m, d)
  const float* y = (const float*)d_in[2];    // (n,)
  const float* W = (const float*)d_in[3];    // (S,)
  const float* lls = (const float*)d_in[4];  // (S,)
  const float* ln = (const float*)d_in[5];   // (1,)

  const int n = in_sizes[2];
  const int S = in_sizes[3];
  const int d = in_sizes[0] / (S * n);
  const int m = in_sizes[1] / (S * d);

  // Workspace layout (floats)
  float* ws = (float*)d_ws;
  float* A = ws;                     // n*n : K -> L (in place, lower)
  float* V = A + (size_t)n * n;      // n*m : Kx^T -> L^-1 Kx^T (in place)
  float* alpha = V + (size_t)n * m;  // n
  float* z = alpha + n;              // n
  float* rnt = z + n;                // S*n row norms (train)
  float* rns = rnt + (size_t)S * n;  // S*m row norms (star)

  gp_rownorm<<<(S * n + 255) / 256, 256, 0, stream>>>(Zt, rnt, S * n, d);
  gp_rownorm<<<(S * m + 255) / 256, 256, 0, stream>>>(Zs, rns, S * m, d);

  // A = K(train,train) + (noise + jitter) I
  gp_build_kernel_mat<<<dim3(n / 16, n / 16), 32, 0, stream>>>(
      Zt, Zt, rnt, rnt, W, lls, ln, 1, S, n, n, d, A);
  // V = Kx^T = K(train, star)  (n x m)
  gp_build_kernel_mat<<<dim3(m / 16, n / 16), 32, 0, stream>>>(
      Zt, Zs, rnt, rns, W, lls, ln, 0, S, n, m, d, V);

  // Blocked right-looking Cholesky (lower), NB = 64
  for (int kb = 0; kb < n; kb += NB) {
    gp_chol_diag<<<1, 256, 0, stream>>>(A, n, kb);
    int rem = n - kb - NB;
    if (rem > 0) {
      gp_chol_trsm<<<(rem + 127) / 128, 128, 0, stream>>>(A, n, kb);
      gp_chol_syrk<<<dim3(rem / 64, rem / 64), 128, 0, stream>>>(A, n, kb);
    }
  }

  // alpha = (L L^T)^-1 y
  gp_trsv_fwd<<<1, 256, 0, stream>>>(A, n, y, z);
  gp_trsv_bwd<<<1, 256, 0, stream>>>(A, n, z, alpha);

  float* mean = (float*)d_out;  // [0, m)
  float* var = mean + m;        // [m, 2m)
  float* mll = var + m;         // [2m]

  // mean = Kx @ alpha (uses V before it is overwritten by the solve)
  gp_mean<<<(m + 255) / 256, 256, 0, stream>>>(V, alpha, mean, n, m);

  // V <- L^-1 V  (blocked forward substitution, WMMA panel updates)
  for (int kb = 0; kb < n; kb += NB) {
    if (kb > 0)
      gp_panel_gemm<<<dim3(m / 64), 128, 0, stream>>>(A, n, V, m, kb);
    gp_panel_trs<<<(m + 127) / 128, 128, 0, stream>>>(A, n, V, m, kb);
  }

  gp_var<<<(m + 255) / 256, 256, 0, stream>>>(V, W, ln, var, n, m, S);
  gp_mll<<<1, 256, 0, stream>>>(A, n, y, alpha, mll);
}